// GCNDropEdge_91087666413880
// MI455X (gfx1250) — compile-verified
//
#include <hip/hip_runtime.h>
#include <hip/hip_bf16.h>
#include <stddef.h>

// ---------------------------------------------------------------------------
// GCN: 3x (GEMM -> SpMM(sum over in-edges) -> *norm + bias -> ReLU -> dropout)
// GEMM uses CDNA5 V_WMMA_F32_16X16X4_F32 (exact fp32, matches reference
// numerics). SpMM = wave-per-edge scatter with non-returning f32 atomics.
// ---------------------------------------------------------------------------

typedef float v2f __attribute__((ext_vector_type(2)));
typedef float v8f __attribute__((ext_vector_type(8)));

#define NNODES 100000
#define NEDGES 1600000
#define KDIM   128
#define HIDDEN 128
#define NCLS   40

// ---------------- zero / degree / norm helpers ----------------

__global__ void zero_f32(float* __restrict__ p, int n) {
    int i = blockIdx.x * blockDim.x + threadIdx.x;
    if (i < n) p[i] = 0.0f;
}

__global__ void degree_kernel(const int* __restrict__ dst, float* __restrict__ deg, int e) {
    int i = blockIdx.x * blockDim.x + threadIdx.x;
    if (i < e) atomicAdd(&deg[dst[i]], 1.0f);
}

__global__ void invert_deg_kernel(float* __restrict__ norm, int n) {
    int i = blockIdx.x * blockDim.x + threadIdx.x;
    if (i < n) norm[i] = 1.0f / fmaxf(norm[i], 1.0f);
}

// ---------------- WMMA fp32 GEMM:  Out[M x N_OUT] = A[M x 128] @ W[128 x N_OUT]
// One wave per 16x16 output tile. K loop: 32 x V_WMMA_F32_16X16X4_F32.
// A-frag (16x4): lanes 0-15 row M=lane, v0/v1 = K{k0,k0+1}; lanes 16-31 = K{k0+2,k0+3}.
// B-frag (4x16): v0 = row k0 (lanes 0-15) / row k0+2 (lanes 16-31); v1 = k0+1 / k0+3.
// C/D (16x16):   VGPR r -> row r (lanes 0-15) / row r+8 (lanes 16-31), col = lane&15.
//
// Out-of-range columns (N_OUT=40 case): load column is CLAMPED (in-bounds,
// branch-free); garbage only affects D columns that are never stored. EXEC
// stays all-ones for the whole WMMA loop (ISA requirement).

template<int N_OUT>
__global__ void __launch_bounds__(128)
wmma_gemm_f32(const float* __restrict__ A,
              const float* __restrict__ W,
              float* __restrict__ Out,
              int col_tiles) {
    const int lane = threadIdx.x & 31;
    const int lo   = lane & 15;
    const int hi   = lane >> 4;           // 0 or 1

    const int colTile = blockIdx.y * blockDim.y + threadIdx.y;
    if (colTile >= col_tiles) return;     // wave-uniform: EXEC stays all-ones

    const int rowBase = blockIdx.x * 16;  // N is an exact multiple of 16
    const int col     = colTile * 16 + lo;
    const int colC    = (col < N_OUT) ? col : (N_OUT - 1);   // clamped, branch-free

    const float* aRow = A + (size_t)(rowBase + lo) * KDIM;   // per-lane A base
    const float* wCol = W + colC;                             // per-lane B base

    v8f c = {};
#pragma unroll
    for (int k0 = 0; k0 < KDIM; k0 += 4) {
        const int k = k0 + 2 * hi;
        v2f a;
        a.x = aRow[k];
        a.y = aRow[k + 1];
        v2f b;
        b.x = wCol[(size_t)k * N_OUT];          // compile-time immediate offsets
        b.y = wCol[(size_t)(k + 1) * N_OUT];
        c = __builtin_amdgcn_wmma_f32_16x16x4_f32(
                /*neg_a=*/false, a, /*neg_b=*/false, b,
                /*c_mod=*/(short)0, c, /*reuse_a=*/false, /*reuse_b=*/false);
    }

    if (col < N_OUT) {
        const int row0 = rowBase + 8 * hi;
#pragma unroll
        for (int r = 0; r < 8; ++r)
            Out[(size_t)(row0 + r) * N_OUT + col] = c[r];
    }
}

// ---------------- SpMM scatter: Agg[dst] += H[src], wave per edge ----------

__global__ void __launch_bounds__(256)
spmm_scatter(const float* __restrict__ H,
             const int* __restrict__ src,
             const int* __restrict__ dst,
             float* __restrict__ Agg,
             int ncol, int e) {
    const int wavesPerBlk = blockDim.x >> 5;
    const int edge = blockIdx.x * wavesPerBlk + (threadIdx.x >> 5);
    const int lane = threadIdx.x & 31;
    if (edge >= e) return;
    const int s = src[edge];
    const int d = dst[edge];
    const float* rs = H   + (size_t)s * ncol;
    float*       rd = Agg + (size_t)d * ncol;
    for (int cidx = lane; cidx < ncol; cidx += 32)
        atomicAdd(&rd[cidx], rs[cidx]);   // non-returning global_atomic_add_f32
}

// ---------------- fused norm * agg + bias -> relu -> dropout ----------------

__global__ void __launch_bounds__(256)
post_kernel(const float* __restrict__ Agg,
            const float* __restrict__ norm,
            const float* __restrict__ bias,
            const unsigned char* __restrict__ mask,  // may be null
            float* __restrict__ Out,
            int ncol, int total, int do_relu) {
    int idx = blockIdx.x * blockDim.x + threadIdx.x;
    if (idx >= total) return;
    const int i = idx / ncol;
    const int cidx = idx - i * ncol;
    float v = Agg[idx] * norm[i] + bias[cidx];
    if (do_relu) v = fmaxf(v, 0.0f);
    if (mask) v = mask[idx] ? v * 2.0f : 0.0f;   // 1/KEEP = 2.0
    Out[idx] = v;
}

// ---------------------------------------------------------------------------

extern "C" void kernel_launch(void* const* d_in, const int* in_sizes, int n_in,
                              void* d_out, int out_size, void* d_ws, size_t ws_size,
                              hipStream_t stream) {
    const float* feat  = (const float*)d_in[0];          // [N, 128]
    const int*   src   = (const int*)d_in[1];            // [E]
    const int*   dst   = (const int*)d_in[2];            // [E]
    const unsigned char* mask1 = (const unsigned char*)d_in[3];  // [N,128] bool
    const unsigned char* mask2 = (const unsigned char*)d_in[4];  // [N,128] bool
    const float* w0 = (const float*)d_in[5];             // [128,128]
    const float* b0 = (const float*)d_in[6];             // [128]
    const float* w1 = (const float*)d_in[7];             // [128,128]
    const float* b1 = (const float*)d_in[8];             // [128]
    const float* w2 = (const float*)d_in[9];             // [128,40]
    const float* b2 = (const float*)d_in[10];            // [40]
    float* out = (float*)d_out;                          // [N,40]

    // workspace layout (floats): norm[N] | bufA[N*128] | bufB[N*128]
    float* norm = (float*)d_ws;
    float* bufA = norm + NNODES;            // 400000 B offset, 16B aligned
    float* bufB = bufA + (size_t)NNODES * HIDDEN;

    const int NH = NNODES * HIDDEN;         // 12.8M
    const int NC = NNODES * NCLS;           // 4.0M
    const dim3 gemmBlk(32, 4);
    const dim3 gemmGridH((NNODES / 16), 2); // 8 col tiles / 4 waves
    const dim3 gemmGridC((NNODES / 16), 1); // 3 col tiles / 4 waves
    const int edgeBlocks = (NEDGES + 7) / 8;           // 8 waves (edges) per block

    // ---- degree / norm ----
    zero_f32<<<(NNODES + 255) / 256, 256, 0, stream>>>(norm, NNODES);
    degree_kernel<<<(NEDGES + 255) / 256, 256, 0, stream>>>(dst, norm, NEDGES);
    invert_deg_kernel<<<(NNODES + 255) / 256, 256, 0, stream>>>(norm, NNODES);

    // ---- layer 0: feat @ w0 -> A; scatter A->B; post B->A (relu, mask1) ----
    wmma_gemm_f32<HIDDEN><<<gemmGridH, gemmBlk, 0, stream>>>(feat, w0, bufA, 8);
    zero_f32<<<(NH + 255) / 256, 256, 0, stream>>>(bufB, NH);
    spmm_scatter<<<edgeBlocks, 256, 0, stream>>>(bufA, src, dst, bufB, HIDDEN, NEDGES);
    post_kernel<<<(NH + 255) / 256, 256, 0, stream>>>(bufB, norm, b0, mask1, bufA,
                                                      HIDDEN, NH, /*relu=*/1);

    // ---- layer 1: A @ w1 -> B; scatter B->A; post A->B (relu, mask2) ----
    wmma_gemm_f32<HIDDEN><<<gemmGridH, gemmBlk, 0, stream>>>(bufA, w1, bufB, 8);
    zero_f32<<<(NH + 255) / 256, 256, 0, stream>>>(bufA, NH);
    spmm_scatter<<<edgeBlocks, 256, 0, stream>>>(bufB, src, dst, bufA, HIDDEN, NEDGES);
    post_kernel<<<(NH + 255) / 256, 256, 0, stream>>>(bufA, norm, b1, mask2, bufB,
                                                      HIDDEN, NH, /*relu=*/1);

    // ---- layer 2: B @ w2 -> A (N x 40); scatter A->B; post B->out ----
    wmma_gemm_f32<NCLS><<<gemmGridC, gemmBlk, 0, stream>>>(bufB, w2, bufA, 3);
    zero_f32<<<(NC + 255) / 256, 256, 0, stream>>>(bufB, NC);
    spmm_scatter<<<edgeBlocks, 256, 0, stream>>>(bufA, src, dst, bufB, NCLS, NEDGES);
    post_kernel<<<(NC + 255) / 256, 256, 0, stream>>>(bufB, norm, b2, /*mask=*/nullptr,
                                                      out, NCLS, NC, /*relu=*/0);
}